// EXTD_81810537054901
// MI455X (gfx1250) — compile-verified
//
#include <hip/hip_runtime.h>
#include <cstdint>
#include <cstddef>

// ---------------- problem constants (match reference) ----------------
#define NUMC        21
#define TOPK        200
#define NMS_TOPK    1000
#define NPAD        1024
#define NMS_T       0.3f
#define CONF_T      0.01f
#define BATCH       32
#define NPRIOR      32768
#define NTHREADS    1024

typedef unsigned int       u32;
typedef unsigned long long u64;
typedef unsigned int u32x4 __attribute__((ext_vector_type(4)));
typedef int          i32x8 __attribute__((ext_vector_type(8)));
typedef int          i32x4 __attribute__((ext_vector_type(4)));

// ---------------- dynamic-LDS layout (bytes) -------------------------
// score[32768] u32 (128KB) is reused as the 1000x32-word NMS bitmask.
// hist[2048] u32 (8KB) is reused as keys[1024] u64 (8KB).
#define OFF_SCORE 0          // 131072 B, later reused as M (128000 B)
#define OFF_HIST  131072     // 8192 B, later reused as keys
#define OFF_MISC  139264     // 64 B   (counters / broadcast slots)
#define OFF_VALS  139328     // 4096 B
#define OFF_X1    143424     // 4096 B
#define OFF_Y1    147520     // 4096 B
#define OFF_X2    151616     // 4096 B
#define OFF_Y2    155712     // 4096 B
#define OFF_AREA  159808     // 4096 B
#define OFF_KEEP  163904     // 4096 B
#define LDS_BYTES 167968     // < 320KB per-WGP LDS on CDNA5

__global__ void EXTD_zero_kernel(float* __restrict__ out, int n) {
  int i = blockIdx.x * blockDim.x + threadIdx.x;
  if (i < n) out[i] = 0.0f;
}

__global__ __launch_bounds__(NTHREADS, 1)
void EXTD_detpost_kernel(const float* __restrict__ loc,
                         const float* __restrict__ conf,
                         const float* __restrict__ prior,
                         float* __restrict__ out) {
  extern __shared__ char smem[];
  u32*   score = (u32*)(smem + OFF_SCORE);
  u32*   hist  = (u32*)(smem + OFF_HIST);
  u64*   keys  = (u64*)(smem + OFF_HIST);   // reuse after histogram passes
  u32*   misc  = (u32*)(smem + OFF_MISC);
  float* vls   = (float*)(smem + OFF_VALS);
  float* bx1   = (float*)(smem + OFF_X1);
  float* by1   = (float*)(smem + OFF_Y1);
  float* bx2   = (float*)(smem + OFF_X2);
  float* by2   = (float*)(smem + OFF_Y2);
  float* bar   = (float*)(smem + OFF_AREA);
  u32*   keep  = (u32*)(smem + OFF_KEEP);
  u32*   Mrow  = (u32*)(smem + OFF_SCORE);  // reuse score region for bitmask

  const int tid = threadIdx.x;
  const int b   = blockIdx.y;       // batch
  const int cls = blockIdx.x + 1;   // foreground class 1..20

  // ============ Stage 1: stage this (b,cls) score column into LDS =========
#if __has_builtin(__builtin_amdgcn_tensor_load_to_lds)
  if (tid < 32) {
    // Tensor Data Mover: 2-D tile, 32768 rows x 1 element (4B), row stride
    // = 21 elements. DMA writes 128KB of contiguous scores to LDS offset 0.
    unsigned long long gaddr =
        (unsigned long long)(uintptr_t)(conf) +
        (((unsigned long long)b * NPRIOR * NUMC) + (unsigned long long)cls) * 4ull;
    u32x4 g0;
    g0[0] = 1u;                                     // count=1, user mode
    g0[1] = 0u;                                     // lds_addr = 0 (score region)
    g0[2] = (u32)(gaddr & 0xFFFFFFFFull);           // global_addr[31:0]
    g0[3] = (u32)((gaddr >> 32) & 0x1FFFFFFull)     // global_addr[56:32]
          | (2u << 30);                             // type = 2 ("image")
    i32x8 g1;
    g1[0] = (int)(2u << 16);                        // data_size=4B; wg_mask=0
    g1[1] = (int)((NUMC & 0xFFFFu) << 16);          // tensor_dim0 lo16 (=21)
    g1[2] = (int)(((NUMC >> 16) & 0xFFFFu)          // tensor_dim0 hi16
          | ((NPRIOR & 0xFFFFu) << 16));            // tensor_dim1 lo16 (=0x8000)
    g1[3] = (int)(((NPRIOR >> 16) & 0xFFFFu)        // tensor_dim1 hi16
          | (1u << 16));                            // tile_dim0 = 1
    g1[4] = (int)(NPRIOR & 0xFFFFu);                // tile_dim1 = 32768; tile_dim2=0
    g1[5] = (int)NUMC;                              // tensor_dim0_stride lo32 (=21)
    g1[6] = 0;                                      // stride hi / dim1_stride lo
    g1[7] = 0;
    i32x4 g2 = {0, 0, 0, 0};                        // dims 2/3 unused (zero)
    i32x4 g3 = {0, 0, 0, 0};                        // dim 4 unused (zero)
    i32x8 g4 = {0, 0, 0, 0, 0, 0, 0, 0};            // extra group (clang-23 form)
    __builtin_amdgcn_tensor_load_to_lds(g0, g1, g2, g3, g4, 0);
    __builtin_amdgcn_s_wait_tensorcnt(0);
  }
  __syncthreads();  // publish DMA'd scores to all 32 waves
  for (int i = tid; i < NPRIOR; i += NTHREADS) {
    float f = __uint_as_float(score[i]);
    score[i] = (f > CONF_T) ? __float_as_uint(f) : 0u;   // mask; bits monotonic
  }
#else
  for (int i = tid; i < NPRIOR; i += NTHREADS) {
    const float* p = &conf[((size_t)b * NPRIOR + (size_t)i) * NUMC + cls];
    __builtin_prefetch(p + NTHREADS * NUMC, 0, 1);
    float f = *p;
    score[i] = (f > CONF_T) ? __float_as_uint(f) : 0u;
  }
#endif
  __syncthreads();

  // ============ Stage 2: exact top-1000 selection via radix histogram =====
  // 47-bit key = (valbits << 15) | (32767 - idx): all keys distinct, order
  // (value desc, index asc) == jax.lax.top_k tie-breaking.
  static constexpr int SH[5] = {36, 25, 14, 3, 0};
  static constexpr int NB[5] = {11, 11, 11, 11, 3};
  u64 prefVal   = 0;
  int prefShift = 47;
  u32 r         = NMS_TOPK;   // rank (1-based from top) still to locate
#pragma unroll
  for (int pass = 0; pass < 5; ++pass) {
    hist[tid] = 0u; hist[tid + 1024] = 0u;
    __syncthreads();
    const int sh  = SH[pass];
    const u32 msk = (1u << NB[pass]) - 1u;
    for (int i = tid; i < NPRIOR; i += NTHREADS) {
      u64 key = ((u64)score[i] << 15) | (u32)(NPRIOR - 1 - i);
      if ((key >> prefShift) == prefVal)
        atomicAdd(&hist[(u32)(key >> sh) & msk], 1u);
    }
    __syncthreads();
    // in-place Hillis-Steele suffix sum: hist[i] = sum_{j>=i} count[j]
    for (int s = 1; s < 2048; s <<= 1) {
      u32 v0 = hist[tid];
      u32 v1 = hist[tid + 1024];
      u32 n0 = (tid + s < 2048) ? hist[tid + s] : 0u;
      u32 n1 = (tid + 1024 + s < 2048) ? hist[tid + 1024 + s] : 0u;
      __syncthreads();
      hist[tid] = v0 + n0;
      hist[tid + 1024] = v1 + n1;
      __syncthreads();
    }
    // unique bucket where cumulative count (from top) crosses r
    for (int z = 0; z < 2; ++z) {
      int i  = tid + z * 1024;
      u32 ci = hist[i];
      u32 cn = (i + 1 < 2048) ? hist[i + 1] : 0u;
      if (ci >= r && cn < r) { misc[1] = (u32)i; misc[2] = r - cn; }
    }
    __syncthreads();
    prefVal   = (prefVal << NB[pass]) | (u64)misc[1];
    r         = misc[2];
    prefShift = sh;
    __syncthreads();
  }
  const u64 T = prefVal;   // exact 1000th-largest key; #(key>=T) == 1000

  // ============ Stage 3: compact the 1000 selected keys, bitonic sort =====
  if (tid == 0) misc[0] = 0u;
  keys[tid] = 0ull;                       // pad slots sort to the end
  __syncthreads();
  for (int i = tid; i < NPRIOR; i += NTHREADS) {
    u64 key = ((u64)score[i] << 15) | (u32)(NPRIOR - 1 - i);
    if (key >= T) {
      u32 pos = atomicAdd(&misc[0], 1u);
      if (pos < NPAD) keys[pos] = key;
    }
  }
  __syncthreads();
  for (int k = 2; k <= NPAD; k <<= 1) {
    for (int j = k >> 1; j > 0; j >>= 1) {
      int i = tid, ixj = i ^ j;
      if (ixj > i) {
        u64 a = keys[i], c = keys[ixj];
        bool desc = ((i & k) == 0);       // overall descending sort
        if (desc ? (a < c) : (a > c)) { keys[i] = c; keys[ixj] = a; }
      }
      __syncthreads();
    }
  }

  // ============ Stage 4: decode boxes only for the survivors ==============
  {
    u64 key = keys[tid];
    int idx = (NPRIOR - 1) - (int)(key & 0x7FFFull);
    float val = __uint_as_float((u32)(key >> 15));
    float4 lc = reinterpret_cast<const float4*>(loc)[(size_t)b * NPRIOR + idx];
    float4 pr = reinterpret_cast<const float4*>(prior)[idx];
    float cx = pr.x + lc.x * 0.1f * pr.z;
    float cy = pr.y + lc.y * 0.1f * pr.w;
    float w  = pr.z * expf(lc.z * 0.2f);
    float h  = pr.w * expf(lc.w * 0.2f);
    float x1 = cx - w * 0.5f, y1 = cy - h * 0.5f;
    float x2 = x1 + w,        y2 = y1 + h;
    vls[tid] = val;
    bx1[tid] = x1; by1[tid] = y1; bx2[tid] = x2; by2[tid] = y2;
    bar[tid] = (x2 - x1) * (y2 - y1);
  }
  __syncthreads();   // score region dead from here on -> reuse as Mrow

  // ============ Stage 5: 1000x1000 IoU bitmask (upper triangular) =========
  for (int w = tid; w < NMS_TOPK * 32; w += NTHREADS) {
    int k = w >> 5;            // row (potential suppressor)
    int c = w & 31;            // 32-column word
    u32 bits = 0u;
    if (c * 32 + 31 > k) {     // word entirely at/below diagonal -> zero
      float kx1 = bx1[k], ky1 = by1[k], kx2 = bx2[k], ky2 = by2[k], ka = bar[k];
      for (int t = 0; t < 32; ++t) {
        int j = c * 32 + t;
        if (j > k && j < NMS_TOPK) {
          float xx1 = fmaxf(bx1[j], kx1);
          float yy1 = fmaxf(by1[j], ky1);
          float xx2 = fminf(bx2[j], kx2);
          float yy2 = fminf(by2[j], ky2);
          float iw = fmaxf(xx2 - xx1, 0.0f);
          float ih = fmaxf(yy2 - yy1, 0.0f);
          float inter = iw * ih;
          // iou > T  <=>  inter > T*(areaJ+areaK-inter)  (positive union)
          if (inter > NMS_T * (bar[j] + ka - inter)) bits |= (1u << t);
        }
      }
    }
    Mrow[w] = bits;
  }
  __syncthreads();

  // ============ Stage 6: serial greedy resolve on one wave ================
  if (tid < 32) {
    const int lane = tid;
    u32 removed = 0u;
    for (int k = 0; k < NMS_TOPK; ++k) {
      u32 rw = (u32)__shfl((int)removed, k >> 5, 32);  // broadcast word w/ bit k
      bool notsup = ((rw >> (k & 31)) & 1u) == 0u;
      bool valid  = vls[k] > CONF_T;
      u32 kept = (valid && notsup) ? 1u : 0u;
      if (kept) removed |= Mrow[k * 32 + lane];
      if (lane == 0) keep[k] = kept;
    }
  }
  __syncthreads();

  // ============ Stage 7: rank kept boxes, emit top-200 entries ============
  if (tid < 32) {
    const int lane = tid;
    int running = 0;
    float* obase = out + (((size_t)b * NUMC + (size_t)cls) * TOPK) * 5;
    for (int base = 0; base < NMS_TOPK; base += 32) {
      int k = base + lane;
      bool kp = (k < NMS_TOPK) && (keep[k] != 0u);
      u32 m32 = (u32)__ballot(kp ? 1 : 0);
      int rank = running + __popc(m32 & ((1u << lane) - 1u));
      if (kp && rank < TOPK) {
        float* o = obase + (size_t)rank * 5;
        o[0] = vls[k]; o[1] = bx1[k]; o[2] = by1[k]; o[3] = bx2[k]; o[4] = by2[k];
      }
      running += __popc(m32);
    }
  }
}

extern "C" void kernel_launch(void* const* d_in, const int* in_sizes, int n_in,
                              void* d_out, int out_size, void* d_ws, size_t ws_size,
                              hipStream_t stream) {
  const float* loc   = (const float*)d_in[0];  // (32, 32768, 4)
  const float* conf  = (const float*)d_in[1];  // (32*32768, 21)
  const float* prior = (const float*)d_in[2];  // (32768, 4)
  float* out = (float*)d_out;                  // (32, 21, 200, 5)
  (void)d_ws; (void)ws_size; (void)in_sizes; (void)n_in;

  // Output is sparse-written; zero it (class 0 and unused slots stay 0).
  EXTD_zero_kernel<<<(out_size + NTHREADS - 1) / NTHREADS, NTHREADS, 0, stream>>>(
      out, out_size);

  // One 1024-thread workgroup per (batch, foreground-class) pair.
  dim3 grid(NUMC - 1, BATCH);
  EXTD_detpost_kernel<<<grid, NTHREADS, LDS_BYTES, stream>>>(loc, conf, prior, out);
}